// MaskedTargets_50629074485706
// MI455X (gfx1250) — compile-verified
//
#include <hip/hip_runtime.h>
#include <hip/hip_bf16.h>

typedef __attribute__((ext_vector_type(2))) float v2f;
typedef __attribute__((ext_vector_type(8))) float v8f;

// Problem constants (match reference exactly)
constexpr int  BB   = 16;          // batch
constexpr int  NPIX = 262144;      // pixels per image
constexpr int  C    = 150;         // n_classes
constexpr int  P    = 4096;        // n_pred_ids
constexpr int  T    = BB * C;      // 2400 distinct target ids
constexpr int  CPAD = 160;         // C padded to 10 WMMA tiles

// Workspace layout in 32-bit words
constexpr long long O_OFF   = 0;                              // u32 [P*T]   (39.3 MB)
constexpr long long NP_OFF  = O_OFF + (long long)P * T;       // f32 [P]
constexpr long long NT_OFF  = NP_OFF + P;                     // f32 [T]
constexpr long long IPC_OFF = NT_OFF + T;                     // f32 [P*CPAD] zero-padded
constexpr long long TP_OFF  = IPC_OFF + (long long)P * CPAD;  // f32 [CPAD*CPAD] zero-padded
constexpr long long WS_WORDS = TP_OFF + (long long)CPAD * CPAD; // = 10,517,856 (mult. of 4)

__global__ void k_zero(uint4* ws, long long n4) {
    long long i = (long long)blockIdx.x * blockDim.x + threadIdx.x;
    long long stride = (long long)gridDim.x * blockDim.x;
    uint4 z = {0u, 0u, 0u, 0u};
    for (; i < n4; i += stride) ws[i] = z;
}

// Co-occurrence histogram: O[p, b*C + cls] += 1  (u32 atomics -> deterministic)
__global__ void k_hist(const int* __restrict__ pred, const int* __restrict__ tgt,
                       unsigned* __restrict__ O) {
    long long n = (long long)BB * NPIX;
    long long i = (long long)blockIdx.x * blockDim.x + threadIdx.x;
    long long stride = (long long)gridDim.x * blockDim.x;
    for (; i < n; i += stride) {
        int p = pred[i];
        int cls = tgt[i];
        int b = (int)(i >> 18);            // i / NPIX (NPIX = 2^18)
        atomicAdd(&O[(long long)p * T + b * C + cls], 1u);
    }
}

// n_pred[p] = sum_t O[p,t]  (one block per row, coalesced)
__global__ void k_rowsum(const unsigned* __restrict__ O, float* __restrict__ npred) {
    int p = blockIdx.x;
    const unsigned* row = O + (long long)p * T;
    unsigned s = 0;
    for (int t = threadIdx.x; t < T; t += blockDim.x) s += row[t];
    __shared__ unsigned sh[256];
    sh[threadIdx.x] = s;
    __syncthreads();
    for (int off = 128; off > 0; off >>= 1) {
        if (threadIdx.x < off) sh[threadIdx.x] += sh[threadIdx.x + off];
        __syncthreads();
    }
    if (threadIdx.x == 0) npred[p] = (float)sh[0];
}

// n_tgt[t] = sum_p O[p,t]  (thread per column, coalesced across t)
__global__ void k_colsum(const unsigned* __restrict__ O, float* __restrict__ ntgt) {
    int t = blockIdx.x * blockDim.x + threadIdx.x;
    if (t >= T) return;
    unsigned s = 0;
    for (int p = 0; p < P; ++p) s += O[(long long)p * T + t];
    ntgt[t] = (float)s;
}

// iou_pc[p,c] = sum_b (O>0 ? O/(n_pred+n_tgt-O) : 0), written with CPAD stride
__global__ void k_iou(const unsigned* __restrict__ O, const float* __restrict__ npred,
                      const float* __restrict__ ntgt, float* __restrict__ ipc) {
    int idx = blockIdx.x * blockDim.x + threadIdx.x;
    if (idx >= P * C) return;
    int p = idx / C, c = idx % C;
    float np = npred[p];
    const unsigned* row = O + (long long)p * T + c;
    float s = 0.0f;
#pragma unroll
    for (int b = 0; b < BB; ++b) {
        unsigned o = row[b * C];
        if (o) {
            float of = (float)o;
            s += of / (np + ntgt[b * C + c] - of);
        }
    }
    ipc[p * CPAD + c] = s;
}

// Copy targets [150x150] into padded [160x160] (pad region pre-zeroed)
__global__ void k_padT(const float* __restrict__ tg, float* __restrict__ tp) {
    int idx = blockIdx.x * blockDim.x + threadIdx.x;
    if (idx >= C * C) return;
    int r = idx / C, c = idx % C;
    tp[r * CPAD + c] = tg[idx];
}

// out[4096,150] = ipc[4096,160] @ targets_pad[160,160] via V_WMMA_F32_16X16X4_F32.
// One wave32 per 16x16 output tile; K marched in steps of 4; EXEC all-ones.
__global__ void k_wmma(const float* __restrict__ A, const float* __restrict__ Bm,
                       float* __restrict__ out) {
    int wave = (int)((blockIdx.x * blockDim.x + threadIdx.x) >> 5);
    int lane = threadIdx.x & 31;
    int nt = wave % (CPAD / 16);
    int mt = wave / (CPAD / 16);

    int l15  = lane & 15;        // A row / B col / D col within tile
    int half = lane >> 4;        // lane group selects K pair
    int koff = half * 2;

    const float* arow = A + (long long)(mt * 16 + l15) * CPAD;

    v8f acc = {};
    for (int k = 0; k < CPAD; k += 4) {
        v2f a, b;
        // A 16x4 frag: VGPR0 = K(koff), VGPR1 = K(koff+1)
        a.x = arow[k + koff];
        a.y = arow[k + koff + 1];
        // B 4x16 frag: rows K(koff), K(koff+1) striped across lanes (col = l15)
        b.x = Bm[(k + koff)     * CPAD + nt * 16 + l15];
        b.y = Bm[(k + koff + 1) * CPAD + nt * 16 + l15];
        acc = __builtin_amdgcn_wmma_f32_16x16x4_f32(
            /*neg_a=*/false, a, /*neg_b=*/false, b,
            /*c_mod=*/(short)0, acc, /*reuse_a=*/false, /*reuse_b=*/false);
    }

    // D layout: VGPR r -> row = r + 8*half, col = l15
#pragma unroll
    for (int r = 0; r < 8; ++r) {
        int gr = mt * 16 + half * 8 + r;
        int gc = nt * 16 + l15;
        if (gc < C) out[gr * C + gc] = acc[r];
    }
}

// Row-normalize out[p, :] /= sum_c out[p, c]
__global__ void k_norm(float* __restrict__ out) {
    int p = blockIdx.x;
    float* row = out + (long long)p * C;
    __shared__ float sh[256];
    float s = 0.0f;
    for (int c = threadIdx.x; c < C; c += blockDim.x) s += row[c];
    sh[threadIdx.x] = s;
    __syncthreads();
    for (int off = 128; off > 0; off >>= 1) {
        if (threadIdx.x < off) sh[threadIdx.x] += sh[threadIdx.x + off];
        __syncthreads();
    }
    float den = sh[0];
    for (int c = threadIdx.x; c < C; c += blockDim.x) row[c] = row[c] / den;
}

extern "C" void kernel_launch(void* const* d_in, const int* in_sizes, int n_in,
                              void* d_out, int out_size, void* d_ws, size_t ws_size,
                              hipStream_t stream) {
    const int*   pred    = (const int*)d_in[0];
    const int*   tgts    = (const int*)d_in[1];
    const float* targets = (const float*)d_in[2];

    unsigned* ws    = (unsigned*)d_ws;
    unsigned* O     = ws + O_OFF;
    float*    npred = (float*)(ws + NP_OFF);
    float*    ntgt  = (float*)(ws + NT_OFF);
    float*    ipc   = (float*)(ws + IPC_OFF);
    float*    tp    = (float*)(ws + TP_OFF);
    float*    out   = (float*)d_out;

    k_zero<<<2048, 256, 0, stream>>>((uint4*)d_ws, WS_WORDS / 4);
    k_hist<<<4096, 256, 0, stream>>>(pred, tgts, O);
    k_rowsum<<<P, 256, 0, stream>>>(O, npred);
    k_colsum<<<(T + 255) / 256, 256, 0, stream>>>(O, ntgt);
    k_iou<<<(P * C + 255) / 256, 256, 0, stream>>>(O, npred, ntgt, ipc);
    k_padT<<<(C * C + 255) / 256, 256, 0, stream>>>(targets, tp);
    // (P/16)*(CPAD/16) = 2560 waves, 4 waves/block -> 640 blocks
    k_wmma<<<(P / 16) * (CPAD / 16) / 4, 128, 0, stream>>>(ipc, tp, out);
    k_norm<<<P, 256, 0, stream>>>(out);
}